// BM25ScoringModel_40329742909606
// MI455X (gfx1250) — compile-verified
//
#include <hip/hip_runtime.h>
#include <hip/hip_bf16.h>
#include <math.h>

typedef __attribute__((ext_vector_type(2))) float v2f;
typedef __attribute__((ext_vector_type(8))) float v8f;

#define BM25_K1 1.2f
#define BM25_K3 8.0f
#define BM25_B  0.75f
#define BM25_N  8841823.0f
#define BM25_LAVE 56.0f

// ---------------------------------------------------------------------------
// Wave32 sum reduction through the matrix pipe:
//   A (16x4 f32, 2 VGPRs/lane) = lane partial duplicated in both components,
//   B (4x16 f32)               = all ones,
//   D = A*B  => sum over all (m,k) of A == 2 * (sum over 32 lanes of s).
// Per the documented C/D layout, lane n holds D[M in its half][N=n] in 8
// VGPRs, so (lane-local sum of 8) + (same from lane n^16) = column sum = 2*S.
// Robust to the exact A/B lane striping; only needs the C/D layout.
// ---------------------------------------------------------------------------
static __device__ __forceinline__ float wave_sum_wmma(float s) {
    v2f a; a.x = s;    a.y = s;
    v2f b; b.x = 1.0f; b.y = 1.0f;
    v8f c = {};
    v8f d = __builtin_amdgcn_wmma_f32_16x16x4_f32(
        /*neg_a=*/false, a, /*neg_b=*/false, b,
        /*c_mod=*/(short)0, c, /*reuse_a=*/false, /*reuse_b=*/false);
    float t = ((d[0] + d[1]) + (d[2] + d[3])) + ((d[4] + d[5]) + (d[6] + d[7]));
    t += __shfl_xor(t, 16, 32);
    return 0.5f * t;
}

// All 256 threads (8 wave32s) must reach this convergently (EXEC all ones).
static __device__ __forceinline__ float block_sum(float acc, float* sm) {
    float w = wave_sum_wmma(acc);
    const int lane = threadIdx.x & 31;
    const int wid  = threadIdx.x >> 5;
    if (lane == 0) sm[wid] = w;
    __syncthreads();
    float t = 0.0f;
    if (threadIdx.x == 0) {
        const int nw = blockDim.x >> 5;
        for (int i = 0; i < nw; ++i) t += sm[i];
    }
    return t;  // valid in thread 0 only
}

// ---------------------------------------------------------------------------
__global__ void bm25_zero_kernel(unsigned int* __restrict__ ws, int n) {
    int i = blockIdx.x * blockDim.x + threadIdx.x;
    const int stride = gridDim.x * blockDim.x;
    for (; i < n; i += stride) ws[i] = 0u;
}

// Scatter-add histogram: first L ids -> hq, second L ids -> hp.
__global__ void bm25_hist_kernel(const int* __restrict__ ids,
                                 unsigned int* __restrict__ hq,
                                 unsigned int* __restrict__ hp,
                                 int L) {
    const int4* __restrict__ ids4 = (const int4*)ids;
    const int n4  = (2 * L) >> 2;
    const int lq4 = L >> 2;
    const int stride = gridDim.x * blockDim.x;
    int i = blockIdx.x * blockDim.x + threadIdx.x;
    for (; i < n4; i += stride) {
        __builtin_prefetch(&ids4[i + stride], 0, 1);  // global_prefetch_b8
        const int4 v = ids4[i];
        unsigned int* __restrict__ h = (i < lq4) ? hq : hp;
        atomicAdd(&h[v.x], 1u);
        atomicAdd(&h[v.y], 1u);
        atomicAdd(&h[v.z], 1u);
        atomicAdd(&h[v.w], 1u);
    }
}

// Gather + BM25 term + two-stage deterministic reduction (stage 1).
__global__ void bm25_score_kernel(const int* __restrict__ q,
                                  const unsigned int* __restrict__ hq,
                                  const unsigned int* __restrict__ hp,
                                  const float* __restrict__ DF,
                                  float* __restrict__ partials,
                                  int L, float denom_c) {
    const int4* __restrict__ q4 = (const int4*)q;
    const int n4 = L >> 2;
    const int stride = gridDim.x * blockDim.x;
    float acc = 0.0f;
    for (int i = blockIdx.x * blockDim.x + threadIdx.x; i < n4; i += stride) {
        __builtin_prefetch(&q4[i + stride], 0, 1);
        const int4 v = q4[i];
        const int id[4] = {v.x, v.y, v.z, v.w};
#pragma unroll
        for (int j = 0; j < 4; ++j) {
            const int t = id[j];
            const float qtf = (float)hq[t];
            const float ptf = (float)hp[t];
            const float df  = DF[t];
            const float idf = __log2f((BM25_N - df + 0.5f) / (df + 0.5f));
            acc += (qtf / (BM25_K3 + qtf)) *
                   (BM25_K1 * ptf / (ptf + denom_c)) * idf;
        }
    }
    __shared__ float sm[8];
    const float t = block_sum(acc, sm);
    if (threadIdx.x == 0) partials[blockIdx.x] = t;
}

// Stage 2: single block sums the fixed-order partials, applies sigmoid.
__global__ void bm25_finalize_kernel(const float* __restrict__ partials,
                                     int nb, float* __restrict__ out) {
    float acc = 0.0f;
    for (int i = threadIdx.x; i < nb; i += blockDim.x) acc += partials[i];
    __shared__ float sm[8];
    const float s = block_sum(acc, sm);
    if (threadIdx.x == 0) out[0] = 1.0f / (1.0f + __expf(-s));
}

// ---------------------------------------------------------------------------
extern "C" void kernel_launch(void* const* d_in, const int* in_sizes, int n_in,
                              void* d_out, int out_size, void* d_ws, size_t ws_size,
                              hipStream_t stream) {
    (void)n_in; (void)out_size; (void)ws_size;
    const int* ids  = (const int*)d_in[0];      // [2, L] int32
    const float* DF = (const float*)d_in[2];    // [VOCAB] float32
    const int L     = in_sizes[0] / 2;
    const int VOCAB = in_sizes[2];

    // Workspace layout: hq[VOCAB] u32 | hp[VOCAB] u32 | partials[NB] f32
    unsigned int* hq = (unsigned int*)d_ws;
    unsigned int* hp = hq + VOCAB;
    const int NB = 1024;                         // stage-1 blocks / partials
    float* partials = (float*)(hp + VOCAB);

    const int BS = 256;                          // 8 wave32s per block

    // 1) zero histograms + partials (workspace is poisoned, must init)
    const int nzero = 2 * VOCAB + NB;
    bm25_zero_kernel<<<2048, BS, 0, stream>>>((unsigned int*)d_ws, nzero);

    // 2) histograms via L2-resident u32 atomics
    bm25_hist_kernel<<<4096, BS, 0, stream>>>(ids, hq, hp, L);

    // 3) gather + score -> NB partials (deterministic mapping)
    const float denom_c =
        BM25_K1 * (1.0f - BM25_B + BM25_B * (float)L / BM25_LAVE);
    bm25_score_kernel<<<NB, BS, 0, stream>>>(ids, hq, hp, DF, partials,
                                             L, denom_c);

    // 4) fixed-order final reduction + sigmoid
    bm25_finalize_kernel<<<1, BS, 0, stream>>>(partials, NB, (float*)d_out);
}